// Int4Quantizer_66254165508541
// MI455X (gfx1250) — compile-verified
//
#include <hip/hip_runtime.h>
#include <stdint.h>

// Problem constants (from reference): x is [8, 4096, 1024] fp32, row-major,
// channel = flat_index % 1024.  N = 32768 rows per channel.
#define C_CH   1024
#define N_ROWS 32768
#define KSEL   32669          // int(99.7 * 32768 / 100), clamped to [1, N-1]
#define NBINS  256
#define CH_TILE 16

typedef __attribute__((ext_vector_type(4))) unsigned int u32x4;
typedef __attribute__((ext_vector_type(8))) int          i32x8;
typedef __attribute__((ext_vector_type(4))) int          i32x4;
typedef __attribute__((ext_vector_type(4))) float        f32x4;

// ---------------------------------------------------------------------------
// Init: zero the global per-channel histograms, reset per-channel radix state.
// ---------------------------------------------------------------------------
__global__ void k_init(uint32_t* __restrict__ hist,
                       uint32_t* __restrict__ prefix,
                       uint32_t* __restrict__ kk) {
  int gid = blockIdx.x * blockDim.x + threadIdx.x;
  if (gid < C_CH * NBINS) hist[gid] = 0u;
  if (gid < C_CH) { prefix[gid] = 0u; kk[gid] = KSEL; }
}

// ---------------------------------------------------------------------------
// Round 1: per-channel histogram over the exponent byte (bits 30:23 of |x|).
// LDS-aggregated: each block owns 16 channels x 256 bins (16 KB), processes a
// 1024-row slab, then flushes nonzero counters to the global histogram.
// ---------------------------------------------------------------------------
__global__ void k_hist_round1(const float* __restrict__ x,
                              uint32_t* __restrict__ hist) {
  __shared__ uint32_t lh[CH_TILE * NBINS];
  for (int i = threadIdx.x; i < CH_TILE * NBINS; i += blockDim.x) lh[i] = 0u;
  __syncthreads();

  const int c0   = blockIdx.x * CH_TILE;
  const int rpb  = N_ROWS / gridDim.y;        // rows per block
  const int r0   = blockIdx.y * rpb;
  const int chL  = threadIdx.x & (CH_TILE - 1);
  const int rOff = threadIdx.x / CH_TILE;     // 0..15

  for (int r = r0 + rOff; r < r0 + rpb; r += 256 / CH_TILE) {
    uint32_t u = __float_as_uint(x[(size_t)r * C_CH + c0 + chL]) & 0x7fffffffu;
    atomicAdd(&lh[chL * NBINS + (u >> 23)], 1u);   // ds_add_u32
  }
  __syncthreads();

  for (int i = threadIdx.x; i < CH_TILE * NBINS; i += blockDim.x) {
    uint32_t v = lh[i];
    if (v) atomicAdd(&hist[(size_t)(c0 + (i >> 8)) * NBINS + (i & 255)], v);
  }
}

// ---------------------------------------------------------------------------
// Rounds 2..4: scan x (L2-resident, 134 MB < 192 MB L2); only elements whose
// high bits match the channel's accumulated prefix are histogrammed (a tiny
// and shrinking fraction), so direct global atomics are cheap.
// ---------------------------------------------------------------------------
__global__ void k_refine(const float* __restrict__ x,
                         const uint32_t* __restrict__ prefix,
                         uint32_t* __restrict__ hist,
                         int shift, int dbits) {
  const uint32_t mask   = (1u << dbits) - 1u;
  const size_t   nvec   = (size_t)N_ROWS * C_CH / 4;
  const size_t   stride = (size_t)gridDim.x * blockDim.x;
  const f32x4*   x4     = (const f32x4*)x;

  for (size_t i = (size_t)blockIdx.x * blockDim.x + threadIdx.x; i < nvec; i += stride) {
    if (i + stride < nvec) __builtin_prefetch(&x4[i + stride], 0, 1);  // global_prefetch_b8
    const f32x4 v  = x4[i];
    const int   cb = (int)((i * 4) & (C_CH - 1));
#pragma unroll
    for (int j = 0; j < 4; ++j) {
      uint32_t u = __float_as_uint(v[j]) & 0x7fffffffu;
      int      c = cb + j;
      if ((u >> (shift + dbits)) == prefix[c])
        atomicAdd(&hist[(size_t)c * NBINS + ((u >> shift) & mask)], 1u);
    }
  }
}

// ---------------------------------------------------------------------------
// Per-channel digit selection: walk the bins, extend prefix, rebase k,
// zero the bins for the next round.  Last round emits scale = value / 7.
// ---------------------------------------------------------------------------
__global__ void k_select(uint32_t* __restrict__ hist,
                         uint32_t* __restrict__ prefix,
                         uint32_t* __restrict__ kk,
                         float* __restrict__ scale,
                         int dbits, int is_last) {
  const int c = blockIdx.x;
  uint32_t* h = hist + (size_t)c * NBINS;
  if (threadIdx.x == 0) {
    const int nb   = 1 << dbits;
    uint32_t  need = kk[c];
    uint32_t  cum  = 0;
    int       dig  = nb - 1;
    for (int d = 0; d < nb; ++d) {
      uint32_t cnt = h[d];
      if (cum + cnt >= need) { dig = d; break; }
      cum += cnt;
    }
    uint32_t p = (prefix[c] << dbits) | (uint32_t)dig;
    prefix[c] = p;
    kk[c]     = need - cum;
    if (is_last) scale[c] = __uint_as_float(p) * (1.0f / 7.0f);
  }
  __syncthreads();
  for (int i = threadIdx.x; i < NBINS; i += blockDim.x) h[i] = 0u;
}

// ---------------------------------------------------------------------------
// Fused stochastic int4 quantize + dequant.  Streams x and rand (float4,
// fully coalesced).  The 4 KB per-channel scale table is DMA'd into LDS via
// the Tensor Data Mover (wave 0 issues, s_wait_tensorcnt 0 fences).
// ---------------------------------------------------------------------------
__global__ void k_quant(const float* __restrict__ x,
                        const float* __restrict__ rnd,
                        const float* __restrict__ g_scale,
                        float* __restrict__ out) {
  __shared__ float s_scale[C_CH];

#if defined(__HIP_DEVICE_COMPILE__) && defined(__gfx1250__)
  if (threadIdx.x < 32) {
    // Tensor DMA descriptor (ISA cdna5 ch.8): 1024x1 tile of 4-byte elements.
    uint64_t ga  = (uint64_t)(uintptr_t)g_scale;
    uint32_t lds = (uint32_t)(uintptr_t)&s_scale[0];  // low 32 bits = LDS offset
    u32x4 g0 = { 1u,                                   // count=1, valid, user mode
                 lds,                                  // lds_addr
                 (uint32_t)ga,                         // global_addr[31:0]
                 (uint32_t)((ga >> 32) & 0x01ffffffu) | (2u << 30) }; // addr[56:32] | type=2
    i32x8 g1 = { (int)(2u << 16),        // data_size = 4 bytes; wg_mask=0
                 (int)(1024u << 16),     // tensor_dim0[15:0] in [31:16]
                 (int)(1u << 16),        // tensor_dim1 = 1
                 (int)(1024u << 16),     // tile_dim0 = 1024
                 (int)1,                 // tile_dim1 = 1, tile_dim2 = 0
                 (int)1024,              // tensor_dim0_stride[31:0]
                 0, 0 };
    i32x4 gz4 = { 0, 0, 0, 0 };
#if __clang_major__ >= 23
    i32x8 gz8 = { 0, 0, 0, 0, 0, 0, 0, 0 };
    __builtin_amdgcn_tensor_load_to_lds(g0, g1, gz4, gz4, gz8, 0);
#else
    __builtin_amdgcn_tensor_load_to_lds(g0, g1, gz4, gz4, 0);
#endif
    __builtin_amdgcn_s_wait_tensorcnt(0);
  }
  __syncthreads();
#else
  for (int i = threadIdx.x; i < C_CH; i += blockDim.x) s_scale[i] = g_scale[i];
  __syncthreads();
#endif

  const f32x4* x4 = (const f32x4*)x;
  const f32x4* r4 = (const f32x4*)rnd;
  f32x4*       o4 = (f32x4*)out;
  const size_t nvec   = (size_t)N_ROWS * C_CH / 4;
  const size_t stride = (size_t)gridDim.x * blockDim.x;

  for (size_t i = (size_t)blockIdx.x * blockDim.x + threadIdx.x; i < nvec; i += stride) {
    if (i + stride < nvec) {
      __builtin_prefetch(&x4[i + stride], 0, 1);   // global_prefetch_b8
      __builtin_prefetch(&r4[i + stride], 0, 1);
    }
    f32x4 xv = x4[i];
    f32x4 rv = r4[i];
    f32x4 ov;
    const int cb = (int)((i * 4) & (C_CH - 1));
#pragma unroll
    for (int j = 0; j < 4; ++j) {
      float sc = s_scale[cb + j];
      float xs = xv[j] / (sc + 1e-8f);          // x / (scale + EPS)
      float fl = floorf(xs);
      float pr = xs - fl;
      float xr = (rv[j] < pr) ? (fl + 1.0f) : fl;
      float xc = fminf(7.0f, fmaxf(-7.0f, xr)); // clip(-L, L)
      ov[j] = xc * sc;                          // dequant (scale without EPS)
    }
    o4[i] = ov;
  }
}

// ---------------------------------------------------------------------------
extern "C" void kernel_launch(void* const* d_in, const int* in_sizes, int n_in,
                              void* d_out, int out_size, void* d_ws, size_t ws_size,
                              hipStream_t stream) {
  const float* x   = (const float*)d_in[0];
  const float* rnd = (const float*)d_in[1];
  float*       out = (float*)d_out;

  // Workspace layout (~1.02 MB):
  uint32_t* hist   = (uint32_t*)d_ws;            // C_CH * NBINS u32 = 1 MB
  uint32_t* prefix = hist + (size_t)C_CH * NBINS;
  uint32_t* kk     = prefix + C_CH;
  float*    scale  = (float*)(kk + C_CH);

  k_init<<<(C_CH * NBINS) / 256, 256, 0, stream>>>(hist, prefix, kk);

  // Round 1: exponent byte, LDS-aggregated histogram.
  dim3 g1(C_CH / CH_TILE, 32);
  k_hist_round1<<<g1, 256, 0, stream>>>(x, hist);
  k_select<<<C_CH, 64, 0, stream>>>(hist, prefix, kk, scale, 8, 0);

  // Rounds 2..4: mantissa refinement (prefix-filtered, L2-resident scans).
  k_refine<<<4096, 256, 0, stream>>>(x, prefix, hist, 15, 8);
  k_select<<<C_CH, 64, 0, stream>>>(hist, prefix, kk, scale, 8, 0);

  k_refine<<<4096, 256, 0, stream>>>(x, prefix, hist, 7, 8);
  k_select<<<C_CH, 64, 0, stream>>>(hist, prefix, kk, scale, 8, 0);

  k_refine<<<4096, 256, 0, stream>>>(x, prefix, hist, 0, 7);
  k_select<<<C_CH, 64, 0, stream>>>(hist, prefix, kk, scale, 7, 1);

  // Fused quantize/dequant (HBM-bound: ~402 MB -> ~17 us at 23.3 TB/s).
  k_quant<<<4096, 256, 0, stream>>>(x, rnd, scale, out);
}